// JaggedLogSoftmaxModule_49314814492715
// MI455X (gfx1250) — compile-verified
//
#include <hip/hip_runtime.h>
#include <math.h>

#define BLOCK   256
#define NWAVES  (BLOCK / 32)
#define LDS_CAP 12288   // fp32 capacity staged per segment (48 KB + skew pad)

__device__ __forceinline__ unsigned ldsAddr(const void* p) {
    // LDS aperture puts the wave-relative byte offset in addr[31:0]
    return (unsigned)(uintptr_t)p;
}

__device__ __forceinline__ float waveReduceMax(float v) {
#pragma unroll
    for (int off = 16; off > 0; off >>= 1)
        v = fmaxf(v, __shfl_xor(v, off, 32));
    return v;
}

__device__ __forceinline__ float waveReduceSum(float v) {
#pragma unroll
    for (int off = 16; off > 0; off >>= 1)
        v += __shfl_xor(v, off, 32);
    return v;
}

__device__ __forceinline__ float blockReduceMax(float v, float* red) {
    const int lane = threadIdx.x & 31, wid = threadIdx.x >> 5;
    v = waveReduceMax(v);
    if (lane == 0) red[wid] = v;
    __syncthreads();
    if (threadIdx.x < 32) {
        float x = (lane < NWAVES) ? red[lane] : -INFINITY;
        x = waveReduceMax(x);
        if (lane == 0) red[0] = x;
    }
    __syncthreads();
    float r = red[0];
    __syncthreads();
    return r;
}

__device__ __forceinline__ float blockReduceSum(float v, float* red) {
    const int lane = threadIdx.x & 31, wid = threadIdx.x >> 5;
    v = waveReduceSum(v);
    if (lane == 0) red[wid] = v;
    __syncthreads();
    if (threadIdx.x < 32) {
        float x = (lane < NWAVES) ? red[lane] : 0.0f;
        x = waveReduceSum(x);
        if (lane == 0) red[0] = x;
    }
    __syncthreads();
    float r = red[0];
    __syncthreads();
    return r;
}

__global__ __launch_bounds__(BLOCK) void jagged_logsoftmax_kernel(
        const float* __restrict__ logits,
        const int*   __restrict__ prefix,
        float*       __restrict__ out,
        int nsegs) {
    __shared__ __align__(16) float smem[LDS_CAP + 4];   // +4 for skew
    __shared__ float red[NWAVES];

    const int seg = blockIdx.x;
    if (seg >= nsegs) return;
    const int start = (seg == 0) ? 0 : prefix[seg - 1];
    const int end   = prefix[seg];
    const int len   = end - start;
    if (len <= 0) return;

    const int tid = threadIdx.x;
    const float* __restrict__ gbase = logits + start;
    float* __restrict__ obase = out + start;

    if (len <= LDS_CAP) {
        // Skewed staging: element i -> smem[skew+i]; after `lead` scalar
        // elements, global and LDS body addresses are both 16B aligned.
        const int skew = start & 3;
        int lead = (4 - skew) & 3;
        if (lead > len) lead = len;
        const int body       = (len - lead) & ~3;   // multiple of 4
        const int nvec       = body >> 2;
        const int tail_start = lead + body;

        // ---- DMA lead (b32) ----
        if (tid < lead) {
            asm volatile("global_load_async_to_lds_b32 %0, %1, %2"
                         :
                         : "v"(ldsAddr(&smem[skew + tid])),
                           "v"((unsigned)tid * 4u), "s"(gbase)
                         : "memory");
        }
        // ---- DMA body (b128: 4 floats per async op) ----
        for (int c = tid; c < nvec; c += BLOCK) {
            const int e = lead + (c << 2);
            asm volatile("global_load_async_to_lds_b128 %0, %1, %2"
                         :
                         : "v"(ldsAddr(&smem[skew + e])),
                           "v"((unsigned)e * 4u), "s"(gbase)
                         : "memory");
        }
        // ---- DMA tail (b32) ----
        for (int i = tail_start + tid; i < len; i += BLOCK) {
            asm volatile("global_load_async_to_lds_b32 %0, %1, %2"
                         :
                         : "v"(ldsAddr(&smem[skew + i])),
                           "v"((unsigned)i * 4u), "s"(gbase)
                         : "memory");
        }
        __builtin_amdgcn_s_wait_asynccnt(0);   // this wave's DMA complete
        __syncthreads();                        // all waves' tiles visible

        // ---- Pass 1: segment max (vectorized LDS reads) ----
        float m = -INFINITY;
        if (tid < lead) m = smem[skew + tid];
        for (int c = tid; c < nvec; c += BLOCK) {
            const float4 v = *(const float4*)&smem[skew + lead + (c << 2)];
            m = fmaxf(m, fmaxf(fmaxf(v.x, v.y), fmaxf(v.z, v.w)));
        }
        for (int i = tail_start + tid; i < len; i += BLOCK)
            m = fmaxf(m, smem[skew + i]);
        m = blockReduceMax(m, red);

        // ---- Pass 2: sum of exp ----
        float s = 0.0f;
        if (tid < lead) s = __expf(smem[skew + tid] - m);
        for (int c = tid; c < nvec; c += BLOCK) {
            const float4 v = *(const float4*)&smem[skew + lead + (c << 2)];
            s += __expf(v.x - m) + __expf(v.y - m) +
                 __expf(v.z - m) + __expf(v.w - m);
        }
        for (int i = tail_start + tid; i < len; i += BLOCK)
            s += __expf(smem[skew + i] - m);
        s = blockReduceSum(s, red);
        const float ls = __logf(s);

        // ---- Pass 3: write output (b128 stores on aligned body) ----
        if (tid < lead) obase[tid] = (smem[skew + tid] - m) - ls;
        for (int c = tid; c < nvec; c += BLOCK) {
            const int e = lead + (c << 2);
            const float4 v = *(const float4*)&smem[skew + e];
            float4 r;
            r.x = (v.x - m) - ls;
            r.y = (v.y - m) - ls;
            r.z = (v.z - m) - ls;
            r.w = (v.w - m) - ls;
            *(float4*)&obase[e] = r;   // 16B aligned: (start+lead)%4 == 0
        }
        for (int i = tail_start + tid; i < len; i += BLOCK)
            obase[i] = (smem[skew + i] - m) - ls;
    } else {
        // ---- Rare slow path (P ~ e^-6): 3 streaming passes, L2-resident ----
        float m = -INFINITY;
        for (int i = tid; i < len; i += BLOCK) m = fmaxf(m, gbase[i]);
        m = blockReduceMax(m, red);

        float s = 0.0f;
        for (int i = tid; i < len; i += BLOCK) s += __expf(gbase[i] - m);
        s = blockReduceSum(s, red);
        const float ls = __logf(s);

        for (int i = tid; i < len; i += BLOCK)
            obase[i] = (gbase[i] - m) - ls;
    }
}

extern "C" void kernel_launch(void* const* d_in, const int* in_sizes, int n_in,
                              void* d_out, int out_size, void* d_ws, size_t ws_size,
                              hipStream_t stream) {
    const float* logits = (const float*)d_in[0];
    const int*   prefix = (const int*)d_in[1];
    float*       out    = (float*)d_out;
    const int nsegs = in_sizes[1];   // one block per segment

    jagged_logsoftmax_kernel<<<nsegs, BLOCK, 0, stream>>>(logits, prefix, out, nsegs);
}